// RGnnLayer_83769042141222
// MI455X (gfx1250) — compile-verified
//
#include <hip/hip_runtime.h>
#include <hip/hip_bf16.h>
#include <stddef.h>

// ---------------------------------------------------------------------------
// R-GCN layer for MI455X (gfx1250, wave32).
//   Pass 1 (WMMA): out = x@W_root + b_root  (written to d_out)
//                  h[r] = x@W_convs[r]      (written to d_ws, 102.4 MB, L2-resident)
//   Pass 2 (scatter): for each relation r, edge (s,d):
//                  out[d] += h[r][s]   via non-returning global_atomic_add_f32
//
// Roofline: GEMMs total 2*100000*64*64*5 ~= 4.1 GFLOP (negligible on WMMA
// units). Scatter moves ~3.3 GB and issues ~410M f32 atomics; h (102.4 MB) and
// out (25.6 MB) both fit in the 192 MB L2, so pass 2 runs at L2/atomic rate,
// not HBM rate.
// ---------------------------------------------------------------------------

typedef float v2f __attribute__((ext_vector_type(2)));
typedef float v8f __attribute__((ext_vector_type(8)));

#define D_DIM      64          // D_IN == D_OUT == 64
#define N_MAT      5           // 4 relation weights + root weight
#define WAVES_PER_BLOCK 8      // 256 threads = 8 wave32

// ---------------------------------------------------------------------------
// Pass 1: tiled GEMM with V_WMMA_F32_16X16X4_F32.
// One wave owns a 16-node tile; loops over the 5 weight matrices, 4 column
// tiles each, K=64 as 16 wmma steps.
//
// f32 WMMA VGPR layouts (ISA 7.12.2, wave32):
//   A 16x4 : lanes 0-15 = rows M (K-half 0), lanes 16-31 = rows M (K-half 1):
//            vgpr0 = A[M][2*kh+0], vgpr1 = A[M][2*kh+1], kh = lane>>4
//   B 4x16 : vgpr0: row K=2*kh+0, N=lane&15 ; vgpr1: row K=2*kh+1
//   C/D    : vgpr i: lanes 0-15 -> M=i, lanes 16-31 -> M=i+8, N=lane&15
//
// LDS weight staging is PRE-SWIZZLED into B-fragment order so each B operand
// is one aligned ds_load_b64 (pair {W[k0][j], W[k0+1][j]} contiguous):
//   sW[ ((kk*2 + kh)*64 + j) * 2 + e ] = W[kk*4 + kh*2 + e][j]
// Column tiles are 128 B apart -> compiler can fuse ds_load_2addr_b64.
// ---------------------------------------------------------------------------
__global__ __launch_bounds__(256) void rgcn_gemm_wmma(
    const float* __restrict__ x,       // [N, 64]
    const float* __restrict__ Wc,      // [4, 64, 64]
    const float* __restrict__ Wr,      // [64, 64]
    const float* __restrict__ br,      // [64]
    float* __restrict__ out,           // [N, 64]
    float* __restrict__ h,             // [4, N, 64]  (workspace)
    int n_nodes)
{
    __shared__ float sW[D_DIM * D_DIM];          // 16 KB, B-fragment swizzled

    const int lane = threadIdx.x & 31;
    const int wave = threadIdx.x >> 5;
    const int tile = blockIdx.x * WAVES_PER_BLOCK + wave;
    const int rc   = lane & 15;                  // A-row / B-col / D-col index
    const int kh   = lane >> 4;                  // K-half select (0 or 1)

    // Wave-uniform predicate -> EXEC stays all-1s inside the WMMA region.
    const bool active = (tile * 16 + 16) <= n_nodes;

    // Preload the 16 A fragments for this node tile (2 VGPRs each = 32 VGPRs).
    v2f a[16];
    if (active) {
        const float* xrow = x + (size_t)(tile * 16 + rc) * D_DIM;
        #pragma unroll
        for (int kk = 0; kk < 16; ++kk)
            a[kk] = *(const v2f*)(xrow + kk * 4 + 2 * kh);
    }

    for (int m = 0; m < N_MAT; ++m) {
        const float* __restrict__ W = (m < 4) ? (Wc + (size_t)m * D_DIM * D_DIM) : Wr;

        __syncthreads();
        #pragma unroll
        for (int i = threadIdx.x; i < D_DIM * D_DIM; i += 256) {
            const int k  = i >> 6;               // source row 0..63
            const int j  = i & 63;               // source col 0..63
            const int kk = k >> 2;
            const int kx = (k >> 1) & 1;         // K-half of the fragment
            const int el = k & 1;                // element within the pair
            sW[((kk * 2 + kx) * D_DIM + j) * 2 + el] = W[i];
        }
        __syncthreads();

        if (!active) continue;                   // wave-uniform

        // Accumulators for the 4 column tiles (4 x 8 VGPRs).
        v8f c[4];
        float* __restrict__ dst;
        if (m == 4) {
            dst = out;                           // root: fold bias into C init
            #pragma unroll
            for (int n = 0; n < 4; ++n) {
                const float bv = br[n * 16 + rc];
                #pragma unroll
                for (int i = 0; i < 8; ++i) c[n][i] = bv;
            }
        } else {
            dst = h + (size_t)m * n_nodes * D_DIM;
            #pragma unroll
            for (int n = 0; n < 4; ++n) {
                #pragma unroll
                for (int i = 0; i < 8; ++i) c[n][i] = 0.0f;
            }
        }

        // Per-lane LDS base for this lane's K-half and column; the kk and n
        // offsets are compile-time-constant byte offsets for ds_load.
        const v2f* __restrict__ bbase =
            (const v2f*)&sW[(kh * D_DIM + rc) * 2];

        #pragma unroll
        for (int kk = 0; kk < 16; ++kk) {
            #pragma unroll
            for (int n = 0; n < 4; ++n) {
                const v2f b = bbase[kk * 2 * D_DIM + n * 16];  // one ds_load_b64
                // (neg_a, A, neg_b, B, c_mod, C, reuse_a, reuse_b)
                c[n] = __builtin_amdgcn_wmma_f32_16x16x4_f32(
                    false, a[kk], false, b, (short)0, c[n], false, false);
            }
        }

        // Store D tiles: vgpr i -> row (i + 8*kh), col = n*16 + rc.
        #pragma unroll
        for (int n = 0; n < 4; ++n) {
            #pragma unroll
            for (int i = 0; i < 8; ++i)
                dst[(size_t)(tile * 16 + i + 8 * kh) * D_DIM + n * 16 + rc] = c[n][i];
        }
    }
}

// ---------------------------------------------------------------------------
// Pass 2: edge scatter. One wave per edge; lane L handles columns L and L+32.
// Gathers hit L2 (h is L2-resident); the non-returning global_atomic_add_f32
// RMWs execute in the L2 atomic units — no HBM round trips in steady state.
// ---------------------------------------------------------------------------
__global__ __launch_bounds__(256) void rgcn_scatter_add(
    const float* __restrict__ h,       // [4, N, 64]
    const int* __restrict__ ei,        // [4, 2, E]
    float* __restrict__ out,           // [N, 64]
    int n_nodes, int n_edges)
{
    const int r = blockIdx.y;
    const long long e = (long long)blockIdx.x * WAVES_PER_BLOCK + (threadIdx.x >> 5);
    if (e >= n_edges) return;

    const int lane = threadIdx.x & 31;
    const int* __restrict__ src_p = ei + (size_t)r * 2 * n_edges;
    const int* __restrict__ dst_p = src_p + n_edges;

    const int s = src_p[e];
    const int d = dst_p[e];

    const float* __restrict__ hrow = h + ((size_t)r * n_nodes + s) * D_DIM;
    float* __restrict__ orow = out + (size_t)d * D_DIM;

    const float v0 = hrow[lane];
    const float v1 = hrow[lane + 32];
    atomicAdd(orow + lane,      v0);
    atomicAdd(orow + lane + 32, v1);
}

extern "C" void kernel_launch(void* const* d_in, const int* in_sizes, int n_in,
                              void* d_out, int out_size, void* d_ws, size_t ws_size,
                              hipStream_t stream) {
    const float* x  = (const float*)d_in[0];   // [N, 64]
    const float* Wc = (const float*)d_in[1];   // [4, 64, 64]
    const float* Wr = (const float*)d_in[2];   // [64, 64]
    const float* br = (const float*)d_in[3];   // [64]
    const int*   ei = (const int*)  d_in[4];   // [4, 2, E]
    float* out = (float*)d_out;
    float* h   = (float*)d_ws;                 // needs 4*N*64*4 B = 102.4 MB

    const int n_nodes = in_sizes[0] / D_DIM;           // 100000
    const int n_edges = in_sizes[4] / (4 * 2);         // 1600000

    const int tiles = (n_nodes + 15) / 16;             // 6250
    dim3 gridA((tiles + WAVES_PER_BLOCK - 1) / WAVES_PER_BLOCK);
    rgcn_gemm_wmma<<<gridA, 256, 0, stream>>>(x, Wc, Wr, br, out, h, n_nodes);

    dim3 gridB((unsigned)((n_edges + WAVES_PER_BLOCK - 1) / WAVES_PER_BLOCK), 4);
    rgcn_scatter_add<<<gridB, 256, 0, stream>>>(h, ei, out, n_nodes, n_edges);
}